// MegaBlocksMoeMLP_80221399155539
// MI455X (gfx1250) — compile-verified
//
#include <hip/hip_runtime.h>
#include <hip/hip_bf16.h>

#define NE      8
#define TOPK    2
#define HID     1024
#define FFN     2048
#define M_TOK   2048
#define TILE_M  32
#define TILES_PER_E (M_TOK / TILE_M)   // 64
#define NCHUNK  1024                   // FFN processed in 2 chunks

// padded LDS row strides (floats): stride % 64 == 4 -> lane-to-lane bank step 4
#define SX_STRIDE (HID + 4)            // 1028
#define SH_STRIDE (NCHUNK + 4)         // 1028

typedef __attribute__((ext_vector_type(2))) float v2f;
typedef __attribute__((ext_vector_type(8))) float v8f;

__device__ __forceinline__ v8f wmma_f32(v2f a, v2f b, v8f c) {
  // D = A(16x4 f32) * B(4x16 f32) + C(16x16 f32)
  return __builtin_amdgcn_wmma_f32_16x16x4_f32(
      /*neg_a=*/false, a, /*neg_b=*/false, b,
      /*c_mod=*/(short)0, c, /*reuse_a=*/false, /*reuse_b=*/false);
}

// CDNA5 async global->LDS copy (16B per lane). LDS byte offset = low 32 bits
// of the generic pointer (LDS aperture occupies addr[63:32]).
__device__ __forceinline__ void async_g2l_b128(unsigned lds_off, const float* gsrc) {
  asm volatile("global_load_async_to_lds_b128 %0, %1, off"
               :: "v"(lds_off), "v"(gsrc) : "memory");
}
__device__ __forceinline__ void wait_asynccnt0() {
  asm volatile("s_wait_asynccnt 0" ::: "memory");
}

// ---------------------------------------------------------------------------
// Zero output accumulator region + per-expert counters
// ---------------------------------------------------------------------------
__global__ __launch_bounds__(256) void moe_zero_kernel(float* __restrict__ out,
                                                       int* __restrict__ count) {
  size_t i = (size_t)blockIdx.x * blockDim.x + threadIdx.x;  // 524288 float4s
  float4 z = make_float4(0.f, 0.f, 0.f, 0.f);
  ((float4*)out)[i] = z;
  if (blockIdx.x == 0 && threadIdx.x < NE) count[threadIdx.x] = 0;
}

// ---------------------------------------------------------------------------
// Router: logits -> softmax -> top2; scatter tokens into per-expert lists
// ---------------------------------------------------------------------------
__global__ __launch_bounds__(256) void moe_router_kernel(
    const float* __restrict__ x, const float* __restrict__ rw,
    const float* __restrict__ rb, float* __restrict__ topk_out,
    int* __restrict__ count, int* __restrict__ tokenlist,
    float* __restrict__ wlist) {
  int m = blockIdx.x * 256 + threadIdx.x;

  float acc[NE];
#pragma unroll
  for (int e = 0; e < NE; ++e) acc[e] = rb[e];

  const float4* x4 = (const float4*)(x + (size_t)m * HID);
#pragma unroll 4
  for (int k4 = 0; k4 < HID / 4; ++k4) {
    float4 xv = x4[k4];
#pragma unroll
    for (int e = 0; e < NE; ++e) {
      float4 wv = ((const float4*)(rw + (size_t)e * HID))[k4];
      acc[e] += xv.x * wv.x + xv.y * wv.y + xv.z * wv.z + xv.w * wv.w;
    }
  }

  // stable softmax over 8
  float mx = acc[0];
#pragma unroll
  for (int e = 1; e < NE; ++e) mx = fmaxf(mx, acc[e]);
  float p[NE], s = 0.f;
#pragma unroll
  for (int e = 0; e < NE; ++e) { p[e] = __expf(acc[e] - mx); s += p[e]; }
  float inv = 1.f / s;
#pragma unroll
  for (int e = 0; e < NE; ++e) p[e] *= inv;

  // top-2 (ties -> lower index, matching lax.top_k)
  int e1 = 0; float p1 = p[0];
#pragma unroll
  for (int e = 1; e < NE; ++e) if (p[e] > p1) { p1 = p[e]; e1 = e; }
  int e2 = (e1 == 0) ? 1 : 0; float p2 = p[e2];
#pragma unroll
  for (int e = 0; e < NE; ++e)
    if (e != e1 && p[e] > p2) { p2 = p[e]; e2 = e; }

  topk_out[m * 2 + 0] = p1;
  topk_out[m * 2 + 1] = p2;

  int pos = atomicAdd(&count[e1], 1);
  tokenlist[e1 * M_TOK + pos] = m;
  wlist[e1 * M_TOK + pos] = p1;
  pos = atomicAdd(&count[e2], 1);
  tokenlist[e2 * M_TOK + pos] = m;
  wlist[e2 * M_TOK + pos] = p2;
}

// ---------------------------------------------------------------------------
// Expert-grouped MoE MLP. 32-token tile per block (halves L2 weight traffic
// vs 16), 8 waves = 2 M-subtiles x 4 N-groups. FFN handled in 2 chunks of
// 1024 so x tile (129KB) + h chunk (129KB) fit CDNA5's 320KB LDS. Partial
// down-GEMM results are atomically combined into the output, weighted by the
// routing probabilities.
// ---------------------------------------------------------------------------
__global__ __launch_bounds__(256, 1) void moe_mlp_kernel(
    const float* __restrict__ x, const float* __restrict__ w1,
    const float* __restrict__ w1b, const float* __restrict__ w2,
    const float* __restrict__ w2b, const int* __restrict__ count,
    const int* __restrict__ tokenlist, const float* __restrict__ wlist,
    float* __restrict__ out) {
  extern __shared__ float smem[];
  float* sx = smem;                         // TILE_M * SX_STRIDE
  float* sh = smem + TILE_M * SX_STRIDE;    // TILE_M * SH_STRIDE
  __shared__ int   stok[TILE_M];
  __shared__ float scw[TILE_M];

  const int e = blockIdx.x / TILES_PER_E;
  const int t = blockIdx.x % TILES_PER_E;
  const int cnt = count[e];
  const int row0 = t * TILE_M;
  if (row0 >= cnt) return;                  // block-uniform exit
  const int nrows = min(TILE_M, cnt - row0);

  const int tid  = threadIdx.x;
  const int lane = tid & 31;
  const int wave = tid >> 5;

  // out-of-range rows clamp to a valid token (EXEC stays uniform for async
  // copies); their results are discarded via scw=0 and the <nrows store guard
  if (tid < TILE_M) {
    int idx = min(tid, nrows - 1);
    stok[tid] = tokenlist[e * M_TOK + row0 + idx];
    scw[tid]  = (tid < nrows) ? wlist[e * M_TOK + row0 + tid] : 0.f;
  }
  __syncthreads();

  // stage x tile (32 x 1024 f32) into LDS with CDNA5 async copies
  {
    unsigned sx_base = (unsigned)(unsigned long long)(void*)sx;
    for (int i = tid; i < TILE_M * (HID / 4); i += 256) {
      int r = i >> 8;           // 256 float4 per row
      int c = i & 255;
      const float* gp = x + (size_t)stok[r] * HID + (size_t)c * 4;
      unsigned lds_off = sx_base + (unsigned)(r * SX_STRIDE + c * 4) * 4u;
      async_g2l_b128(lds_off, gp);
    }
    wait_asynccnt0();
  }
  __syncthreads();

  const int lane15 = lane & 15;         // M (A frag) / N (B frag) index
  const int khalf  = (lane >> 4) << 1;  // K pair: 0 or 2
  const int msub   = wave >> 2;         // M-subtile 0..1
  const int ngrp   = wave & 3;          // N group  0..3
  const int mBase  = (lane >> 4) << 3;  // C/D row base within subtile
  const int nC     = lane15;            // C/D column

  const float* w1e  = w1  + (size_t)e * (2 * FFN) * HID;
  const float* w1be = w1b + (size_t)e * (2 * FFN);
  const float* w2e  = w2  + (size_t)e * HID * FFN;
  const float* w2be = w2b + (size_t)e * HID;
  const float* paX  = sx + (size_t)(msub * 16 + lane15) * SX_STRIDE + khalf;
  const float* paH  = sh + (size_t)(msub * 16 + lane15) * SH_STRIDE + khalf;

  for (int chunk = 0; chunk < FFN / NCHUNK; ++chunk) {
    const int cb = chunk * NCHUNK;
    if (chunk) __syncthreads();  // previous partial down-GEMM done with sh

    // ---- GEMM1 chunk: gu = x @ w1[e]^T, fused bias + SiLU*up -> sh -------
    for (int nt = ngrp; nt < NCHUNK / 16; nt += 4) {
      const int n0 = cb + nt * 16;
      v8f cg = {0.f,0.f,0.f,0.f,0.f,0.f,0.f,0.f};
      v8f cu = {0.f,0.f,0.f,0.f,0.f,0.f,0.f,0.f};
      const float* pg = w1e + (size_t)(n0 + lane15) * HID + khalf;  // gate row
      const float* pu = pg + (size_t)FFN * HID;                     // up row
      __builtin_prefetch(pg, 0, 0);
      __builtin_prefetch(pu, 0, 0);
#pragma unroll 8
      for (int k = 0; k < HID; k += 4) {
        v2f a  = *(const v2f*)(paX + k);
        v2f bg = *(const v2f*)(pg + k);
        v2f bu = *(const v2f*)(pu + k);
        cg = wmma_f32(a, bg, cg);
        cu = wmma_f32(a, bu, cu);
      }
      const float biasg = w1be[n0 + nC];
      const float biasu = w1be[FFN + n0 + nC];
#pragma unroll
      for (int j = 0; j < 8; ++j) {
        float g = cg[j] + biasg;
        float u = cu[j] + biasu;
        float hv = (g / (1.f + __expf(-g))) * u;          // silu(g) * u
        sh[(size_t)(msub * 16 + mBase + j) * SH_STRIDE + nt * 16 + nC] = hv;
      }
    }
    __syncthreads();

    // ---- partial GEMM2: down += h_chunk @ w2[e,:,chunk]^T -----------------
    for (int kt = ngrp; kt < HID / 16; kt += 4) {
      const int kh0 = kt * 16;
      v8f c = {0.f,0.f,0.f,0.f,0.f,0.f,0.f,0.f};
      const float* pb = w2e + (size_t)(kh0 + lane15) * FFN + cb + khalf;
      __builtin_prefetch(pb, 0, 0);
#pragma unroll 8
      for (int n = 0; n < NCHUNK; n += 4) {
        v2f a = *(const v2f*)(paH + n);
        v2f b = *(const v2f*)(pb + n);
        c = wmma_f32(a, b, c);
      }
      const float bias = (chunk == 0) ? w2be[kh0 + nC] : 0.f;  // bias once
#pragma unroll
      for (int j = 0; j < 8; ++j) {
        int mC = msub * 16 + mBase + j;
        if (mC < nrows) {
          float v = (c[j] + bias) * scw[mC];
          atomicAdd(out + (size_t)stok[mC] * HID + kh0 + nC, v);
        }
      }
    }
  }
}

// ---------------------------------------------------------------------------
extern "C" void kernel_launch(void* const* d_in, const int* in_sizes, int n_in,
                              void* d_out, int out_size, void* d_ws, size_t ws_size,
                              hipStream_t stream) {
  const float* x   = (const float*)d_in[0];
  const float* rw  = (const float*)d_in[1];
  const float* rb  = (const float*)d_in[2];
  const float* w1  = (const float*)d_in[3];
  const float* w1b = (const float*)d_in[4];
  const float* w2  = (const float*)d_in[5];
  const float* w2b = (const float*)d_in[6];

  float* out      = (float*)d_out;
  float* topk_out = out + (size_t)M_TOK * HID;   // second tuple output

  int*   count     = (int*)d_ws;
  int*   tokenlist = count + NE;
  float* wlist     = (float*)(tokenlist + NE * M_TOK);

  // zero accumulator region (B*S*HID floats = 524288 float4) + counters
  moe_zero_kernel<<<2048, 256, 0, stream>>>(out, count);

  // router + scatter
  moe_router_kernel<<<M_TOK / 256, 256, 0, stream>>>(x, rw, rb, topk_out,
                                                     count, tokenlist, wlist);

  // expert-grouped MLP
  size_t smem_bytes = (size_t)(TILE_M * SX_STRIDE + TILE_M * SH_STRIDE) * sizeof(float);
  hipFuncSetAttribute((const void*)moe_mlp_kernel,
                      hipFuncAttributeMaxDynamicSharedMemorySize,
                      (int)smem_bytes);
  moe_mlp_kernel<<<NE * TILES_PER_E, 256, smem_bytes, stream>>>(
      x, w1, w1b, w2, w2b, count, tokenlist, wlist, out);
}